// VAE_random_embedding_39625368273137
// MI455X (gfx1250) — compile-verified
//
#include <hip/hip_runtime.h>

typedef __attribute__((ext_vector_type(2))) float v2f;
typedef __attribute__((ext_vector_type(8))) float v8f;

#define DD   128   // embedding dim
#define RR   2     // rank of update
#define NOBJ 9

// 256 threads = 8 waves (wave32). Each wave produces a 16(b) x 128(d) strip
// via 8 x V_WMMA_F32_16X16X4_F32 with C = gathered mu tile (fused bias add).
__global__ __launch_bounds__(256) void vae_embed_wmma(
    const int*   __restrict__ idx,    // [B]
    const float* __restrict__ ns,     // [B, D]
    const float* __restrict__ ortho,  // [D, D]
    const float* __restrict__ mu,     // [NOBJ, D]
    const float* __restrict__ var,    // [NOBJ, R]
    float*       __restrict__ out,    // [B, D]
    int Btot)
{
    __shared__ float lds_mu[NOBJ * DD];   // 4608 B, gathered bias table
    __shared__ float lds_orT[RR * DD];    // orT[r][d] = ortho[d][r], 1 KB

    const int tid = threadIdx.x;

    // ---- stage tiny tables into LDS (block-wide, then barrier) ----
    for (int t = tid; t < NOBJ * DD; t += 256)
        lds_mu[t] = mu[t];
    {
        // exactly 256 = R*D elements: thread t -> (r = t&1, d = t>>1)
        const int r = tid & (RR - 1);
        const int d = tid >> 1;
        lds_orT[r * DD + d] = ortho[d * DD + r];
    }
    __syncthreads();

    const int wave = tid >> 5;
    const int lane = tid & 31;
    const int half = lane >> 4;    // 0: K=0,1 slots ; 1: K=2,3 (zero-pad) slots
    const int l16  = lane & 15;
    const int rbase = blockIdx.x * 128 + wave * 16;
    if (rbase >= Btot) return;     // whole-wave guard (B % 128 == 0 in harness)

    // every lane loads the idx of row (rbase + lane%16): lanes 16-31 mirror 0-15
    const int row   = rbase + l16;
    const int myidx = idx[row];

    // ---- A matrix: 16x4 f32. VGPR0 = K0 (lanes<16) / K2 (lanes>=16),
    //                          VGPR1 = K1 / K3. K2,K3 are the zero padding. ----
    v2f a;
    {
        const float n0 = ns[(size_t)row * DD + 0];
        const float n1 = ns[(size_t)row * DD + 1];
        const float w0 = var[myidx * RR + 0];
        const float w1 = var[myidx * RR + 1];
        const float mask = (half == 0) ? 1.0f : 0.0f;   // v_cndmask, no branch
        a.x = mask * (w0 * w0 * n0);
        a.y = mask * (w1 * w1 * n1);
    }

    // ---- per-output-row mu offsets for the C gather (cross-lane shuffle) ----
    // D/C layout: VGPR v holds (M = v + 8*half, N = lane%16)
    int muoff[8];
#pragma unroll
    for (int v = 0; v < 8; ++v)
        muoff[v] = __shfl(myidx, v + 8 * half, 32) * DD;

    const float bmask = (half == 0) ? 1.0f : 0.0f;

#pragma unroll
    for (int j = 0; j < 8; ++j) {
        const int dbase = j * 16;

        // B matrix 4x16: V0 = K0 row (lanes<16) / K2 (zeros), V1 = K1 / K3
        v2f b;
        b.x = bmask * lds_orT[0 * DD + dbase + l16];
        b.y = bmask * lds_orT[1 * DD + dbase + l16];

        // C = mu[idx[row_M], dbase + N]  (fused bias accumulate)
        v8f c;
#pragma unroll
        for (int v = 0; v < 8; ++v)
            c[v] = lds_mu[muoff[v] + dbase + l16];

        // D = A x B + C   ->  v_wmma_f32_16x16x4_f32
        c = __builtin_amdgcn_wmma_f32_16x16x4_f32(
                /*neg_a=*/false, a, /*neg_b=*/false, b,
                /*c_mod=*/(short)0, c, /*reuse_a=*/false, /*reuse_b=*/false);

        // store: each half-wave writes 64B contiguous per VGPR (coalesced)
#pragma unroll
        for (int v = 0; v < 8; ++v)
            out[(size_t)(rbase + v + 8 * half) * DD + dbase + l16] = c[v];
    }
}

extern "C" void kernel_launch(void* const* d_in, const int* in_sizes, int n_in,
                              void* d_out, int out_size, void* d_ws, size_t ws_size,
                              hipStream_t stream) {
    // setup_inputs order: img[B,4] (unused), random_int[B] (int32),
    // normal_sample[B,D], orthogonal_set[D,D], mu_[NOBJ,D], var_[NOBJ,R]
    const int*   idx   = (const int*)  d_in[1];
    const float* ns    = (const float*)d_in[2];
    const float* ortho = (const float*)d_in[3];
    const float* mu    = (const float*)d_in[4];
    const float* var   = (const float*)d_in[5];
    float* out = (float*)d_out;

    const int Btot = in_sizes[1];          // 131072
    const int grid = (Btot + 127) / 128;   // 128 rows per block

    vae_embed_wmma<<<grid, 256, 0, stream>>>(idx, ns, ortho, mu, var, out, Btot);
}